// SecondOrderConeProjector_29102698397877
// MI455X (gfx1250) — compile-verified
//
#include <hip/hip_runtime.h>

#define SOC_EPS 1e-12f

// Second-order cone projection, one thread per row of D floats.
// Rows are staged global -> LDS with the gfx1250 async-copy path
// (GLOBAL_LOAD_ASYNC_TO_LDS_B128, ASYNCcnt), then consumed via ds_load_b128.
//
// LDS layout is SoA: buf[j][tid] (j = float4 chunk index). A wave-wide
// b128 access then touches a contiguous 512B LDS region -> bank-conflict
// free for every D, on both the async writes and the ds_load reads.
//
// The async instruction adds IOFFSET to BOTH the LDS and the global address
// (ISA §15.18), so chunk j uses the shared global base + offset:16j, with the
// LDS base VGPR pre-biased by j*(4096-16) to land on buf[j][tid].
//
// Each lane reads only the LDS slots it wrote itself, so a wave-local
// s_wait_asynccnt 0 is the only synchronization required (no barrier).
template <int D>
__global__ __launch_bounds__(256) void soc_proj_kernel(const float* __restrict__ x,
                                                       float* __restrict__ out,
                                                       int m) {
    constexpr int VEC = D / 4;                 // float4 chunks per row
    __shared__ float4 buf[VEC][256];           // SoA staging, conflict-free

    const int tid = threadIdx.x;
    const int row = blockIdx.x * 256 + tid;

    // LDS byte address of buf[0][tid] (as3 offset == low 32 bits of flat addr)
    const unsigned lds0 = (unsigned)(unsigned long long)(&buf[0][tid]);
    const float4* g = reinterpret_cast<const float4*>(x) + (long long)row * VEC;

    // Per-chunk LDS stride between buf[j][tid] and buf[j+1][tid] is 4096B;
    // IOFFSET contributes 16B of that, so bias the VGPR by 4096-16 per chunk.
    constexpr unsigned LDS_BIAS = 256u * 16u - 16u;

    if (row < m) {
        asm volatile("global_load_async_to_lds_b128 %0, %1, off"
                     :: "v"(lds0), "v"(g) : "memory");
        if constexpr (VEC > 1)
            asm volatile("global_load_async_to_lds_b128 %0, %1, off offset:16"
                         :: "v"(lds0 + 1u * LDS_BIAS), "v"(g) : "memory");
        if constexpr (VEC > 2) {
            asm volatile("global_load_async_to_lds_b128 %0, %1, off offset:32"
                         :: "v"(lds0 + 2u * LDS_BIAS), "v"(g) : "memory");
            asm volatile("global_load_async_to_lds_b128 %0, %1, off offset:48"
                         :: "v"(lds0 + 3u * LDS_BIAS), "v"(g) : "memory");
        }
    }
    asm volatile("s_wait_asynccnt 0" ::: "memory");
    if (row >= m) return;

    // Pull the row out of LDS (ds_load_b128, conflict-free).
    float v[D];
#pragma unroll
    for (int j = 0; j < VEC; ++j) {
        float4 q = buf[j][tid];
        v[4 * j + 0] = q.x;
        v[4 * j + 1] = q.y;
        v[4 * j + 2] = q.z;
        v[4 * j + 3] = q.w;
    }

    const float t = v[0];
    float ss = 0.0f;
#pragma unroll
    for (int i = 1; i < D; ++i) ss += v[i] * v[i];
    float nz = sqrtf(ss);
    nz = fmaxf(nz, SOC_EPS);

    float o[D];
    if (nz <= t + SOC_EPS) {
        // Inside the cone: identity.
#pragma unroll
        for (int i = 0; i < D; ++i) o[i] = v[i];
    } else if (nz <= -t) {
        // Inside the polar cone: zero.
#pragma unroll
        for (int i = 0; i < D; ++i) o[i] = 0.0f;
    } else {
        // Project onto the cone boundary.
        const float a = 0.5f * (1.0f + t / nz);
        o[0] = a * nz;
#pragma unroll
        for (int i = 1; i < D; ++i) o[i] = a * v[i];
    }

    float4* ov = reinterpret_cast<float4*>(out) + (long long)row * VEC;
#pragma unroll
    for (int j = 0; j < VEC; ++j) {
        float4 q;
        q.x = o[4 * j + 0];
        q.y = o[4 * j + 1];
        q.z = o[4 * j + 2];
        q.w = o[4 * j + 3];
        ov[j] = q;
    }
}

extern "C" void kernel_launch(void* const* d_in, const int* in_sizes, int n_in,
                              void* d_out, int out_size, void* d_ws, size_t ws_size,
                              hipStream_t stream) {
    (void)in_sizes; (void)n_in; (void)out_size; (void)d_ws; (void)ws_size;

    const float* x = (const float*)d_in[0];
    float* out = (float*)d_out;

    // GROUPS = [(4, 2000000), (8, 1000000), (16, 500000)], laid out contiguously.
    const int m4 = 2000000, m8 = 1000000, m16 = 500000;
    size_t off = 0;

    soc_proj_kernel<4><<<(m4 + 255) / 256, 256, 0, stream>>>(x + off, out + off, m4);
    off += (size_t)4 * m4;

    soc_proj_kernel<8><<<(m8 + 255) / 256, 256, 0, stream>>>(x + off, out + off, m8);
    off += (size_t)8 * m8;

    soc_proj_kernel<16><<<(m16 + 255) / 256, 256, 0, stream>>>(x + off, out + off, m16);
}